// GKAN_54992761258608
// MI455X (gfx1250) — compile-verified
//
#include <hip/hip_runtime.h>
#include <hip/hip_bf16.h>
#include <math.h>

#define N_NODES 100000
#define N_EDGES 1600000
#define IN_DIM  500
#define HID     128
#define OUT_DIM 16
#define DEGP1   4
#define N_KAN   2
#define BPAD    16   // B LDS pad (k2-row stride = (128+16)*2 floats == 32 mod 64 banks)

typedef __attribute__((ext_vector_type(2))) float v2f;
typedef __attribute__((ext_vector_type(8))) float v8f;

static __device__ __forceinline__ v8f wmma_f32(v2f a, v2f b, v8f c) {
  // D = A(16x4 f32) x B(4x16 f32) + C(16x16 f32)
  return __builtin_amdgcn_wmma_f32_16x16x4_f32(false, a, false, b, (short)0, c,
                                               false, false);
}

// ---------------------------------------------------------------------------
// Repack cheb_coeffs[l][i][o][d] -> Wk[l][k=4i+d][o]  (K-major GEMM operand)
// ---------------------------------------------------------------------------
__global__ __launch_bounds__(256) void prep_coeffs_kernel(
    const float* __restrict__ C, float* __restrict__ Wk) {
  int idx = blockIdx.x * 256 + threadIdx.x;           // < 2*512*128
  if (idx >= N_KAN * DEGP1 * HID * HID) return;
  int l   = idx / (DEGP1 * HID * HID);
  int rem = idx - l * (DEGP1 * HID * HID);
  int k   = rem / HID;
  int o   = rem - k * HID;
  int i   = k >> 2;
  int d   = k & 3;
  Wk[idx] = C[(((long)l * HID + i) * HID + o) * DEGP1 + d];
}

// ---------------------------------------------------------------------------
// h = x @ W_in : [100000 x 500] * [500 x 128], fp32 WMMA 16x16x4
// block = 128 threads = 4 waves, each wave -> 16 rows x 128 cols.
// B staged pair-interleaved in LDS: Bp[k/2][n*2 + (k&1)] so each lane's
// fragment is a single aligned ds_load_b64 into an even VGPR pair.
// ---------------------------------------------------------------------------
__global__ __launch_bounds__(128) void gemm_in_kernel(
    const float* __restrict__ X, const float* __restrict__ W,
    float* __restrict__ Hout) {
  __shared__ float As[64][36];                 // 64 rows x 32 k (padded)
  __shared__ float Bp[16][(HID + BPAD) * 2];   // 16 k-pairs x 128 n, interleaved

  const int tid  = threadIdx.x;
  const int wave = tid >> 5;
  const int lane = tid & 31;
  const int lr   = lane & 15;
  const int hi   = lane >> 4;         // 0: K0/K1 half, 1: K2/K3 half
  const int m0   = blockIdx.x * 64;

  const v8f vzero = {0.f,0.f,0.f,0.f,0.f,0.f,0.f,0.f};
  v8f acc[8];
#pragma unroll
  for (int i = 0; i < 8; ++i) acc[i] = vzero;

  for (int kt = 0; kt < IN_DIM; kt += 32) {
    // stage A tile (64x32), coalesced float4
#pragma unroll
    for (int it = 0; it < 4; ++it) {
      int idx = it * 128 + tid;                 // 0..511
      int r   = idx >> 3;
      int c4  = (idx & 7) * 4;
      int gr  = m0 + r; if (gr >= N_NODES) gr = N_NODES - 1;
      float4 v = make_float4(0.f, 0.f, 0.f, 0.f);
      if (kt + c4 < IN_DIM) v = *(const float4*)(X + (long)gr * IN_DIM + kt + c4);
      *(float4*)&As[r][c4] = v;
    }
    // stage B tile (32 k x 128 n) pair-interleaved: thread owns (k-pair, n-pair)
#pragma unroll
    for (int it = 0; it < 8; ++it) {
      int idx = it * 128 + tid;                 // 0..1023
      int k2  = idx >> 6;                       // 0..15
      int n   = (idx & 63) * 2;                 // even column
      int gk  = kt + k2 * 2;
      float2 w0 = make_float2(0.f, 0.f), w1 = w0;
      if (gk < IN_DIM) {                        // IN_DIM even -> covers gk+1 too
        w0 = *(const float2*)(W + (long)gk * HID + n);
        w1 = *(const float2*)(W + (long)(gk + 1) * HID + n);
      }
      *(float4*)&Bp[k2][n * 2] = make_float4(w0.x, w1.x, w0.y, w1.y);
    }
    // prefetch next k-tile toward WGP while this tile computes
    if (kt + 32 < IN_DIM) {
      int pr = m0 + (tid >> 1); if (pr >= N_NODES) pr = N_NODES - 1;
      __builtin_prefetch(X + (long)pr * IN_DIM + kt + 32 + (tid & 1) * 16, 0, 3);
      __builtin_prefetch(W + (long)(kt + 32 + (tid >> 4)) * HID + (tid & 15) * 8,
                         0, 3);
    }
    __syncthreads();

    const int ar = wave * 16 + lr;
#pragma unroll
    for (int kk = 0; kk < 32; kk += 4) {
      v2f a;
      a.x = As[ar][kk + 2 * hi + 0];
      a.y = As[ar][kk + 2 * hi + 1];
#pragma unroll
      for (int nt = 0; nt < 8; ++nt) {
        int col = nt * 16 + lr;
        v2f b = *(const v2f*)&Bp[(kk >> 1) + hi][col * 2];
        acc[nt] = wmma_f32(a, b, acc[nt]);
      }
    }
    __syncthreads();
  }

  // C/D layout: lane = column, VGPR j = row (j + 8*hi)
  const int rbase = m0 + wave * 16 + hi * 8;
#pragma unroll
  for (int nt = 0; nt < 8; ++nt) {
    int col = nt * 16 + lr;
#pragma unroll
    for (int j = 0; j < 8; ++j) {
      int gr = rbase + j;
      if (gr < N_NODES) Hout[(long)gr * HID + col] = acc[nt][j];
    }
  }
}

// ---------------------------------------------------------------------------
// Fused ChebKAN: Hout = ChebBasis(tanh(Hin)) @ Wk   (K = 128*4 = 512)
// A-fragment built in-register: lanes 0-15 -> (T0,T1), lanes 16-31 -> (T2,T3)
// ---------------------------------------------------------------------------
__global__ __launch_bounds__(128) void cheb_gemm_kernel(
    const float* __restrict__ Hin, const float* __restrict__ Wk,
    float* __restrict__ Hout) {
  __shared__ float Hs[64][12];                 // 64 rows x 8 i (padded)
  __shared__ float Bp[16][(HID + BPAD) * 2];   // 16 k-pairs x 128 n, interleaved

  const int tid  = threadIdx.x;
  const int wave = tid >> 5;
  const int lane = tid & 31;
  const int lr   = lane & 15;
  const int hi   = lane >> 4;
  const int m0   = blockIdx.x * 64;

  const v8f vzero = {0.f,0.f,0.f,0.f,0.f,0.f,0.f,0.f};
  v8f acc[8];
#pragma unroll
  for (int i = 0; i < 8; ++i) acc[i] = vzero;

  for (int itile = 0; itile < HID / 8; ++itile) {   // 16 tiles of 8 hidden cols
    // stage H tile (64x8)
    {
      int r  = tid >> 1;
      int c4 = (tid & 1) * 4;
      int gr = m0 + r; if (gr >= N_NODES) gr = N_NODES - 1;
      *(float4*)&Hs[r][c4] =
          *(const float4*)(Hin + (long)gr * HID + itile * 8 + c4);
    }
    // stage B tile (32 k x 128 n) pair-interleaved (K = 512 exact, no tail)
#pragma unroll
    for (int it = 0; it < 8; ++it) {
      int idx = it * 128 + tid;
      int k2  = idx >> 6;
      int n   = (idx & 63) * 2;
      int gk  = itile * 32 + k2 * 2;
      float2 w0 = *(const float2*)(Wk + (long)gk * HID + n);
      float2 w1 = *(const float2*)(Wk + (long)(gk + 1) * HID + n);
      *(float4*)&Bp[k2][n * 2] = make_float4(w0.x, w1.x, w0.y, w1.y);
    }
    if (itile + 1 < HID / 8) {
      int pr = m0 + (tid >> 1); if (pr >= N_NODES) pr = N_NODES - 1;
      __builtin_prefetch(Hin + (long)pr * HID + (itile + 1) * 8, 0, 3);
      __builtin_prefetch(
          Wk + (long)((itile + 1) * 32 + (tid >> 2)) * HID + (tid & 3) * 32, 0, 3);
    }
    __syncthreads();

    const int ar = wave * 16 + lr;
#pragma unroll
    for (int ii = 0; ii < 8; ++ii) {
      float t  = tanhf(Hs[ar][ii]);
      float t2 = 2.0f * t * t - 1.0f;        // T2
      float t3 = t * (4.0f * t * t - 3.0f);  // T3
      v2f a;
      a.x = hi ? t2 : 1.0f;                  // (T0,T1) | (T2,T3) via cndmask
      a.y = hi ? t3 : t;
#pragma unroll
      for (int nt = 0; nt < 8; ++nt) {
        int col = nt * 16 + lr;
        v2f b = *(const v2f*)&Bp[2 * ii + hi][col * 2];
        acc[nt] = wmma_f32(a, b, acc[nt]);
      }
    }
    __syncthreads();
  }

  const int rbase = m0 + wave * 16 + hi * 8;
#pragma unroll
  for (int nt = 0; nt < 8; ++nt) {
    int col = nt * 16 + lr;
#pragma unroll
    for (int j = 0; j < 8; ++j) {
      int gr = rbase + j;
      if (gr < N_NODES) Hout[(long)gr * HID + col] = acc[nt][j];
    }
  }
}

// ---------------------------------------------------------------------------
// SpMM scatter: out[row] += h[col] per edge. One wave per edge (32 x float4).
// ---------------------------------------------------------------------------
__global__ __launch_bounds__(256) void spmm_kernel(
    const int* __restrict__ ei, const float* __restrict__ Hin,
    float* __restrict__ Hout) {
  long t = (long)blockIdx.x * 256 + threadIdx.x;
  long e = t >> 5;
  int  c = (int)(t & 31) << 2;
  if (e < N_EDGES) {
    int row = ei[e];
    int col = ei[N_EDGES + e];
    float4 v = *(const float4*)(Hin + (long)col * HID + c);
    float* dst = Hout + (long)row * HID + c;
    atomicAdd(dst + 0, v.x);
    atomicAdd(dst + 1, v.y);
    atomicAdd(dst + 2, v.z);
    atomicAdd(dst + 3, v.w);
  }
}

// ---------------------------------------------------------------------------
// logits = h @ W_out; log_softmax. Half-wave (16 lanes) per node.
// ---------------------------------------------------------------------------
__global__ __launch_bounds__(256) void out_kernel(
    const float* __restrict__ H, const float* __restrict__ Wout,
    float* __restrict__ out) {
  __shared__ float Ws[HID * OUT_DIM];
  for (int i = threadIdx.x; i < HID * OUT_DIM; i += 256) Ws[i] = Wout[i];
  __syncthreads();

  int wave = threadIdx.x >> 5;
  int lane = threadIdx.x & 31;
  int node = (blockIdx.x * 8 + wave) * 2 + (lane >> 4);
  int o    = lane & 15;
  int n    = node < N_NODES ? node : N_NODES - 1;

  const float* hr = H + (long)n * HID;
  float acc = 0.f;
#pragma unroll 8
  for (int k = 0; k < HID; ++k) acc = fmaf(hr[k], Ws[k * OUT_DIM + o], acc);

  float m = acc;
  for (int s = 8; s; s >>= 1) m = fmaxf(m, __shfl_xor(m, s, 16));
  float ex  = expf(acc - m);
  float sum = ex;
  for (int s = 8; s; s >>= 1) sum += __shfl_xor(sum, s, 16);

  if (node < N_NODES) out[(long)node * OUT_DIM + o] = acc - m - logf(sum);
}

// ---------------------------------------------------------------------------
extern "C" void kernel_launch(void* const* d_in, const int* in_sizes, int n_in,
                              void* d_out, int out_size, void* d_ws,
                              size_t ws_size, hipStream_t stream) {
  const float* x    = (const float*)d_in[0];
  const int*   ei   = (const int*)d_in[1];
  const float* Win  = (const float*)d_in[2];
  const float* cheb = (const float*)d_in[3];
  const float* Wout = (const float*)d_in[4];
  float*       out  = (float*)d_out;

  const size_t HN = (size_t)N_NODES * HID;   // 12.8M floats
  float* hA = (float*)d_ws;
  float* hB = hA + HN;
  float* Wk = hB + HN;                       // 2*512*128 floats

  // repack Chebyshev coefficients (K-major)
  prep_coeffs_kernel<<<(N_KAN * DEGP1 * HID * HID + 255) / 256, 256, 0,
                       stream>>>(cheb, Wk);

  const dim3 gGemm((N_NODES + 63) / 64);
  const dim3 gSpmm((unsigned)(((long)N_EDGES * 32 + 255) / 256));

  // h = x @ W_in
  gemm_in_kernel<<<gGemm, 128, 0, stream>>>(x, Win, hA);
  // h = A h
  hipMemsetAsync(hB, 0, HN * sizeof(float), stream);
  spmm_kernel<<<gSpmm, 256, 0, stream>>>(ei, hA, hB);

  for (int l = 0; l < N_KAN; ++l) {
    // h = ChebKAN_l(h)
    cheb_gemm_kernel<<<gGemm, 128, 0, stream>>>(
        hB, Wk + (size_t)l * DEGP1 * HID * HID, hA);
    // h = A h
    hipMemsetAsync(hB, 0, HN * sizeof(float), stream);
    spmm_kernel<<<gSpmm, 256, 0, stream>>>(ei, hA, hB);
  }

  // log_softmax(h @ W_out)
  out_kernel<<<N_NODES / 16, 256, 0, stream>>>(hB, Wout, out);
}